// MyAvgPool2D_56444460204139
// MI455X (gfx1250) — compile-verified
//
#include <hip/hip_runtime.h>
#include <hip/hip_bf16.h>

// Reference reduces to the identity: H=56, OUT_H=56 -> stride=1, k=1, so the
// "avg pool" is a 1x1/stride-1 window: out == x. Pure streaming copy,
// 205.5 MB in + 205.5 MB out => ~17.6 us floor at 23.3 TB/s.
//
// Implementation: stage through LDS with CDNA5 async DMA
// (GLOBAL_LOAD_ASYNC_TO_LDS_B128 / GLOBAL_STORE_ASYNC_FROM_LDS_B128, ASYNCcnt).
// Each block iteration: 8 batched async loads per wave (512B contiguous per
// instruction), one s_wait_asynccnt 0, 8 async stores, one wait. 32KB LDS
// staging buffer per 256-thread block.

#define TPB  256
#define NBUF 8   // float4 buffers per thread per iteration

__global__ __launch_bounds__(TPB) void MyAvgPool2D_identity_async_copy(
    const float4* __restrict__ src, float4* __restrict__ dst,
    size_t n4, size_t n4_main)
{
    __shared__ float4 smem[TPB * NBUF];   // 32 KB staging
    const unsigned tid = threadIdx.x;
    // Low 32 bits of a generic pointer to LDS are the allocation-relative
    // LDS byte address.
    const unsigned lds_base = (unsigned)(uintptr_t)(&smem[0]);

    const size_t blockChunk = (size_t)TPB * NBUF;          // 2048 float4 = 32 KB
    for (size_t base = (size_t)blockIdx.x * blockChunk; base < n4_main;
         base += (size_t)gridDim.x * blockChunk) {

        // Batch 8 async global->LDS loads (deep ASYNCcnt pipeline per wave).
#pragma unroll
        for (int j = 0; j < NBUF; ++j) {
            const float4* p = src + base + (size_t)j * TPB + tid;
            unsigned loff = lds_base + (unsigned)((j * TPB + tid) * sizeof(float4));
            asm volatile("global_load_async_to_lds_b128 %0, %1, off"
                         :: "v"(loff), "v"(p) : "memory");
        }
        asm volatile("s_wait_asynccnt 0x0" ::: "memory");

        // Batch 8 async LDS->global stores (each lane stores its own slot,
        // so no workgroup barrier is required).
#pragma unroll
        for (int j = 0; j < NBUF; ++j) {
            float4* q = dst + base + (size_t)j * TPB + tid;
            unsigned loff = lds_base + (unsigned)((j * TPB + tid) * sizeof(float4));
            asm volatile("global_store_async_from_lds_b128 %0, %1, off"
                         :: "v"(q), "v"(loff) : "memory");
        }
        // Must drain before LDS slots are reused next iteration.
        asm volatile("s_wait_asynccnt 0x0" ::: "memory");
    }

    // float4 tail (dead for this shape: n4 is an exact multiple of 2048).
    const size_t gid      = (size_t)blockIdx.x * TPB + tid;
    const size_t nthreads = (size_t)gridDim.x * TPB;
    for (size_t i = n4_main + gid; i < n4; i += nthreads)
        dst[i] = src[i];
}

// Scalar tail for element counts not divisible by 4 (dead for this shape).
__global__ void MyAvgPool2D_scalar_tail(const float* __restrict__ src,
                                        float* __restrict__ dst,
                                        size_t start, size_t n)
{
    size_t i = start + threadIdx.x;
    if (i < n) dst[i] = src[i];
}

extern "C" void kernel_launch(void* const* d_in, const int* in_sizes, int n_in,
                              void* d_out, int out_size, void* d_ws, size_t ws_size,
                              hipStream_t stream) {
    (void)in_sizes; (void)n_in; (void)d_ws; (void)ws_size;

    const float* x  = (const float*)d_in[0];
    float*      out = (float*)d_out;

    const size_t n  = (size_t)out_size;      // 32*512*56*56 = 51,380,224 floats
    const size_t n4 = n / 4;                 // 12,845,056 float4
    const size_t blockChunk = (size_t)TPB * NBUF;        // 2048 float4
    const size_t n4_main    = (n4 / blockChunk) * blockChunk;

    unsigned grid = (unsigned)((n4 + blockChunk - 1) / blockChunk);  // 6272
    if (grid == 0) grid = 1;

    MyAvgPool2D_identity_async_copy<<<grid, TPB, 0, stream>>>(
        (const float4*)x, (float4*)out, n4, n4_main);

    const size_t rem = n % 4;                // 0 for this shape
    if (rem) {
        MyAvgPool2D_scalar_tail<<<1, 64, 0, stream>>>(x, out, n4 * 4, n);
    }
}